// PillarNet_46394236731910
// MI455X (gfx1250) — compile-verified
//
#include <hip/hip_runtime.h>

// ---------------------------------------------------------------------------
// PillarNet fused inference for MI455X (gfx1250, wave32, WMMA).
//
//  K1 precompute : fold BN into weights/biases (tiny, 64 threads)
//  K2 pointnet   : fp32 WMMA 16x16x4, K=9 padded to 12, max over 100 points.
//                  4 waves/block cooperate on one 16-pillar tile (25 points
//                  each, partial maxes merged in LDS) -> 12000 waves keep
//                  ~16 MB in flight, enough to saturate 23.3 TB/s on the
//                  172.8 MB x stream (NT loads; keep L2 for feat).
//  K3 init       : out <- relu(bias2) per channel (canvas-zero contribution)
//  K4 conv+scatter: 64x64 GEMM on the 48000 occupied pillars only (WMMA,
//                  16 chained K-steps, A-fragments from LDS), ReLU, then
//                  uint-atomicMax into the pooled output (valid: values >= 0).
//
// Roofline: stage 1 reads 172.8 MB -> ~7.4 us at 23.3 TB/s but carries
// 32 FLOP/B; pure fp32 VALU would be ~7x compute-bound, so the matrix pipe is
// mandatory. Stage 3's canvas is 92.5% zeros -> never materialize it: init
// output with the per-channel zero-pixel constant and scatter-max only the
// 48000 real pillars (0.39 GFLOP at 100% WMMA K-efficiency).
// ---------------------------------------------------------------------------

typedef __attribute__((ext_vector_type(2))) float v2f;
typedef __attribute__((ext_vector_type(4))) float v4f;
typedef __attribute__((ext_vector_type(8))) float v8f;

#define B_     4
#define F_     9
#define MP_    100
#define P_     12000
#define C_     64
#define NY_    400
#define NX_    400
#define POOLW_ 100
#define XO_    (NX_ / POOLW_)   // 4
#define KP1    12               // stage-1 K padded 9 -> 12 (3 WMMA k-steps)
#define NTILE  (P_ / 16)        // 750 pillar tiles per batch
#define SPLIT_ 4                // waves cooperating on one tile
#define MPW_   (MP_ / SPLIT_)   // 25 points per wave

// workspace layout in floats
#define WS_W1S  0               // [64][12] BN-folded, zero-padded W1
#define WS_B1   768             // [64] folded bias1
#define WS_W2S  832             // [64][64] BN-folded W2
#define WS_B2   (832 + 4096)    // [64] folded bias2
#define WS_FEAT 4992            // [B][P][64] pillar features (12.3 MB, L2-resident)

// ---------------------------------------------------------------------------
__global__ void precompute_kernel(
    const float* __restrict__ w1, const float* __restrict__ b1,
    const float* __restrict__ g1, const float* __restrict__ be1,
    const float* __restrict__ m1, const float* __restrict__ v1,
    const float* __restrict__ w2, const float* __restrict__ b2,
    const float* __restrict__ g2, const float* __restrict__ be2,
    const float* __restrict__ m2, const float* __restrict__ v2,
    float* __restrict__ ws)
{
    const int c = threadIdx.x;          // 64 threads
    // relu(s*(W x + b) + t) == relu((s*W) x + (s*b + t))
    const float s1 = g1[c] * rsqrtf(v1[c] + 1e-3f);
    const float t1 = be1[c] - m1[c] * s1;
    for (int f = 0; f < KP1; ++f)
        ws[WS_W1S + c * KP1 + f] = (f < F_) ? s1 * w1[c * F_ + f] : 0.0f;
    ws[WS_B1 + c] = s1 * b1[c] + t1;

    const float s2 = g2[c] * rsqrtf(v2[c] + 1e-3f);
    const float t2 = be2[c] - m2[c] * s2;
    for (int k = 0; k < C_; ++k)
        ws[WS_W2S + c * C_ + k] = s2 * w2[c * C_ + k];
    ws[WS_B2 + c] = s2 * b2[c] + t2;
}

// ---------------------------------------------------------------------------
// Stage 1: one block (4 waves) = one 16-pillar tile; wave w covers points
// [w*25, w*25+25). Per mp: 12 f32 WMMAs (4 channel groups x 3 chained K-steps)
// + 32 v_max; next mp's B-fragment is register-prefetched. Partial maxes are
// merged across the 4 waves through LDS (identical lane layout -> plain
// element-wise max), then bias + ReLU applied once and feat stored.
// ---------------------------------------------------------------------------
__global__ __launch_bounds__(128)
void pointnet_kernel(const float* __restrict__ x, float* __restrict__ ws)
{
    __shared__ float red[SPLIT_ - 1][32 * 32];     // waves 1..3 partial maxes

    const int lane = threadIdx.x & 31;
    const int wave = threadIdx.x >> 5;             // segment id 0..3
    const int tile = blockIdx.x;                   // 0 .. 2999
    const int b    = tile / NTILE;
    const int p0   = (tile % NTILE) * 16;
    const int mp0  = wave * MPW_;

    const int half = lane >> 4;                    // 0: K+0/1, 1: K+2/3
    const int lid  = lane & 15;                    // M (A) / N (B,C,D)

    // A fragments from folded W1 (lane l: M=lid, vgpr j: K = kt*4 + 2*half + j)
    const float* __restrict__ w1s = ws + WS_W1S;
    v2f a[4][3];
#pragma unroll
    for (int mg = 0; mg < 4; ++mg)
#pragma unroll
        for (int kt = 0; kt < 3; ++kt)
            a[mg][kt] = *(const v2f*)(w1s + (mg * 16 + lid) * KP1 + kt * 4 + half * 2);

    v8f vmax[4];
#pragma unroll
    for (int mg = 0; mg < 4; ++mg)
#pragma unroll
        for (int r = 0; r < 8; ++r) vmax[mg][r] = -3.4e38f;

    // per-lane base pointer into x; consecutive mp -> += P_
    const float* __restrict__ xp = x + (size_t)b * F_ * MP_ * P_ + p0 + lid;

    // B fragment loader: lane l: N=lid (pillar), vgpr j: K = kt*4 + 2*half + j
    auto load_frag = [&](int mp, v2f bf[3]) {
#pragma unroll
        for (int kt = 0; kt < 3; ++kt)
#pragma unroll
            for (int j = 0; j < 2; ++j) {
                const int f = kt * 4 + half * 2 + j;
                float v = 0.0f;
                if (f < F_)                        // zero-pad K 9..11
                    v = __builtin_nontemporal_load(xp + ((size_t)f * MP_ + mp) * P_);
                bf[kt][j] = v;
            }
    };

    v2f cur[3], nxt[3];
    load_frag(mp0, cur);
    for (int i = 0; i < MPW_; ++i) {
        if (i + 1 < MPW_)                          // prefetch next iteration
            load_frag(mp0 + i + 1, nxt);
#pragma unroll
        for (int mg = 0; mg < 4; ++mg) {
            v8f acc = {};
#pragma unroll
            for (int kt = 0; kt < 3; ++kt)
                acc = __builtin_amdgcn_wmma_f32_16x16x4_f32(
                    false, a[mg][kt], false, cur[kt], (short)0, acc, false, false);
#pragma unroll
            for (int r = 0; r < 8; ++r)
                vmax[mg][r] = fmaxf(vmax[mg][r], acc[r]);
        }
#pragma unroll
        for (int kt = 0; kt < 3; ++kt) cur[kt] = nxt[kt];
    }

    // cross-wave max reduction via LDS (same lane layout in every wave)
    if (wave != 0) {
#pragma unroll
        for (int mg = 0; mg < 4; ++mg)
#pragma unroll
            for (int r = 0; r < 8; ++r)
                red[wave - 1][lane * 32 + mg * 8 + r] = vmax[mg][r];
    }
    __syncthreads();

    if (wave == 0) {
        const float* __restrict__ bias1 = ws + WS_B1;
        float* __restrict__ feat = ws + WS_FEAT;
        const size_t frow = (size_t)(b * P_ + p0 + lid) * C_;
#pragma unroll
        for (int mg = 0; mg < 4; ++mg)
#pragma unroll
            for (int r = 0; r < 8; ++r) {
                float v = vmax[mg][r];
#pragma unroll
                for (int w = 0; w < SPLIT_ - 1; ++w)
                    v = fmaxf(v, red[w][lane * 32 + mg * 8 + r]);
                const int c = mg * 16 + half * 8 + r;   // D layout: M = r + 8*half
                // feat stored RT so it parks in the 192MB L2 for stage 3
                feat[frow + c] = fmaxf(v + bias1[c], 0.0f);
            }
    }
}

// ---------------------------------------------------------------------------
// Stage 3a: empty-canvas contribution — every output cell starts at
// relu(bias2[d]) (what conv+BN+ReLU yields on a zero pixel).
// ---------------------------------------------------------------------------
__global__ __launch_bounds__(256)
void init_out_kernel(const float* __restrict__ ws, float* __restrict__ out)
{
    const int i = blockIdx.x * 256 + threadIdx.x;  // 409600 total
    const int d = (i / (NY_ * XO_)) & (C_ - 1);
    out[i] = fmaxf(ws[WS_B2 + d], 0.0f);
}

// ---------------------------------------------------------------------------
// Stage 3b: GEMM only on occupied pillars + scatter-max into pooled output.
// One wave = 16 pillars x 64 output channels; K=64 via 16 chained WMMAs.
// W2 (16KB) + bias2 staged in LDS once per block; fragments read as ds_load_b64.
// ---------------------------------------------------------------------------
__global__ __launch_bounds__(128)
void scatter_conv_kernel(const int* __restrict__ coords,
                         const float* __restrict__ ws,
                         float* __restrict__ out)
{
    __shared__ float w2s_lds[C_ * C_];
    __shared__ float bias2_lds[C_];

    {   // cooperative stage-in (128 threads x 8 float4 = 16KB)
        const v4f* src = (const v4f*)(ws + WS_W2S);
        v4f* dst = (v4f*)w2s_lds;
#pragma unroll
        for (int i = threadIdx.x; i < (C_ * C_) / 4; i += 128)
            dst[i] = src[i];
        if (threadIdx.x < C_)
            bias2_lds[threadIdx.x] = ws[WS_B2 + threadIdx.x];
    }
    __syncthreads();

    const int lane = threadIdx.x & 31;
    const int wave = threadIdx.x >> 5;
    const int task = blockIdx.x * 4 + wave;        // 0 .. 2999
    const int b    = task / NTILE;
    const int p0   = (task % NTILE) * 16;
    const int half = lane >> 4;
    const int lid  = lane & 15;

    const float* __restrict__ feat = ws + WS_FEAT;

    // B fragments: 16 pillars x K=64 channels (contiguous float2 loads, L2 hits)
    v2f bf[16];
    {
        const float* fp = feat + (size_t)(b * P_ + p0 + lid) * C_;
#pragma unroll
        for (int kt = 0; kt < 16; ++kt)
            bf[kt] = *(const v2f*)(fp + kt * 4 + half * 2);
    }

    const int row = b * P_ + p0 + lid;             // my pillar
    const int y   = coords[row * 4 + 2];
    const int xq  = coords[row * 4 + 3] / POOLW_;
    unsigned int* __restrict__ outu = (unsigned int*)out;

#pragma unroll
    for (int mg = 0; mg < 4; ++mg) {
        v8f acc = {};
#pragma unroll
        for (int kt = 0; kt < 16; ++kt) {
            const v2f av = *(const v2f*)(w2s_lds + (mg * 16 + lid) * C_ + kt * 4 + half * 2);
            acc = __builtin_amdgcn_wmma_f32_16x16x4_f32(
                false, av, false, bf[kt], (short)0, acc, false, false);
        }
#pragma unroll
        for (int r = 0; r < 8; ++r) {
            const int d   = mg * 16 + half * 8 + r;
            const float v = fmaxf(acc[r] + bias2_lds[d], 0.0f);
            const size_t oi = ((size_t)(b * C_ + d) * NY_ + y) * XO_ + xq;
            // all values >= 0 -> float max == uint max on the bit pattern
            atomicMax(outu + oi, __float_as_uint(v));
        }
    }
}

// ---------------------------------------------------------------------------
extern "C" void kernel_launch(void* const* d_in, const int* in_sizes, int n_in,
                              void* d_out, int out_size, void* d_ws, size_t ws_size,
                              hipStream_t stream)
{
    const float* x      = (const float*)d_in[0];
    const int*   coords = (const int*)  d_in[1];
    const float* w1  = (const float*)d_in[2];
    const float* b1  = (const float*)d_in[3];
    const float* g1  = (const float*)d_in[4];
    const float* be1 = (const float*)d_in[5];
    const float* m1  = (const float*)d_in[6];
    const float* v1  = (const float*)d_in[7];
    const float* w2  = (const float*)d_in[8];
    const float* b2  = (const float*)d_in[9];
    const float* g2  = (const float*)d_in[10];
    const float* be2 = (const float*)d_in[11];
    const float* m2  = (const float*)d_in[12];
    const float* v2  = (const float*)d_in[13];
    float* out = (float*)d_out;
    float* ws  = (float*)d_ws;

    precompute_kernel<<<1, 64, 0, stream>>>(w1, b1, g1, be1, m1, v1,
                                            w2, b2, g2, be2, m2, v2, ws);
    pointnet_kernel<<<B_ * NTILE, 128, 0, stream>>>(x, ws);
    init_out_kernel<<<(B_ * C_ * NY_ * XO_) / 256, 256, 0, stream>>>(ws, out);
    scatter_conv_kernel<<<(B_ * NTILE) / 4, 128, 0, stream>>>(coords, ws, out);
}